// FourierContinuationFNO_DarcyL_89489938580017
// MI455X (gfx1250) — compile-verified
//
#include <hip/hip_runtime.h>
#include <math.h>

// ---------------------------------------------------------------------------
// FNO (Darcy) forward on MI455X / gfx1250.
// All FFTs are truncated-mode DFTs -> dense fp32 GEMMs on V_WMMA_F32_16X16X4_F32.
// Roofline: ~24 GFLOP total vs ~1.6 GB HBM traffic -> memory bound; fp32 kept.
// ---------------------------------------------------------------------------

typedef float v2f __attribute__((ext_vector_type(2)));
typedef float v8f __attribute__((ext_vector_type(8)));

static __device__ __forceinline__ v8f wmma_f32(v2f a, v2f b, v8f c) {
    // D = A(16x4) x B(4x16) + C(16x16), fp32, wave32
    return __builtin_amdgcn_wmma_f32_16x16x4_f32(
        /*neg_a=*/false, a, /*neg_b=*/false, b,
        /*c_mod=*/(short)0, c, /*reuse_a=*/false, /*reuse_b=*/false);
}

// Problem constants
constexpr int NB  = 16;     // batch
constexpr int SS  = 256;    // spatial
constexpr int DV  = 32;     // channels
constexpr int NL  = 4;      // layers
constexpr int PIX = SS * SS;          // 65536
constexpr int BC  = NB * DV;          // 512
constexpr int NROW = BC * SS;         // 131072 (b,c,h) rows

// Workspace layout (float offsets)
constexpr size_t OFF_H0  = 0;
constexpr size_t SZ_H    = (size_t)BC * PIX;          // 33,554,432
constexpr size_t OFF_H1  = OFF_H0 + SZ_H;
constexpr size_t SZ_T1   = (size_t)BC * SS * 16;      // 2,097,152
constexpr size_t OFF_T1R = OFF_H1 + SZ_H;
constexpr size_t OFF_T1I = OFF_T1R + SZ_T1;
constexpr size_t SZ_T2   = (size_t)BC * 32 * 16;      // 262,144
constexpr size_t OFF_T2R = OFF_T1I + SZ_T1;
constexpr size_t OFF_T2I = OFF_T2R + SZ_T2;
constexpr size_t SZ_T3   = (size_t)BC * 12 * 16 * 2;  // 196,608
constexpr size_t OFF_T3R = OFF_T2I + SZ_T2;
constexpr size_t OFF_T3I = OFF_T3R + SZ_T3;
constexpr size_t SZ_T4   = (size_t)BC * SS * 16;      // 2,097,152
constexpr size_t OFF_T4R = OFF_T3I + SZ_T3;
constexpr size_t OFF_T4I = OFF_T4R + SZ_T4;
constexpr size_t OFF_FPR = OFF_T4I + SZ_T4;           // [128][16][2] pair-interleaved
constexpr size_t OFF_FPI = OFF_FPR + 4096;
constexpr size_t OFF_GHR = OFF_FPI + 4096;            // [32][256] row-major (rows 24..31 zero)
constexpr size_t OFF_GHI = OFF_GHR + 8192;
constexpr size_t OFF_IGR = OFF_GHI + 8192;            // [256][24] row-major
constexpr size_t OFF_IGI = OFF_IGR + 6144;
constexpr size_t OFF_ICR = OFF_IGI + 6144;            // [8][256][2] pair-interleaved
constexpr size_t OFF_ICI = OFF_ICR + 4096;

// ---------------------------------------------------------------------------
// Basis tables (phase reduced mod 256 for accuracy).
// ---------------------------------------------------------------------------
__global__ void build_basis(float* ws) {
    const float STEP = 6.28318530717958647692f / 256.0f;
    int t = threadIdx.x;   // 0..255
    float* Fpr = ws + OFF_FPR;  float* Fpi = ws + OFF_FPI;
    float* Ghr = ws + OFF_GHR;  float* Ghi = ws + OFF_GHI;
    float* IGr = ws + OFF_IGR;  float* IGi = ws + OFF_IGI;
    float* ICr = ws + OFF_ICR;  float* ICi = ws + OFF_ICI;

    // Stage A B-matrix: B[w][ky] = e^{-2*pi*i*ky*w/256}, ky>=12 zero. Pair over w.
    for (int ky = 0; ky < 16; ++ky) {
        int ph = (ky * t) & 255;
        float th = (float)ph * STEP;
        float cr = (ky < 12) ? cosf(th) : 0.0f;
        float ci = (ky < 12) ? -sinf(th) : 0.0f;
        int idx = ((t >> 1) * 16 + ky) * 2 + (t & 1);
        Fpr[idx] = cr; Fpi[idx] = ci;
    }
    // Stage B A-matrix: Gh[kx_row][h] = e^{-2*pi*i*kx*h/256}; rows 24..31 zero.
    for (int r = 0; r < 32; ++r) {
        int kx = (r < 12) ? r : ((r < 24) ? (232 + r) : 0);  // 0..11, 244..255
        int ph = (kx * t) & 255;
        float th = (float)ph * STEP;
        Ghr[r * 256 + t] = (r < 24) ? cosf(th) : 0.0f;
        Ghi[r * 256 + t] = (r < 24) ? -sinf(th) : 0.0f;
    }
    // Stage C A-matrix: IGh[h][kx] = e^{+2*pi*i*kx*h/256} / 256.
    for (int c = 0; c < 24; ++c) {
        int kx = (c < 12) ? c : (232 + c);
        int ph = (kx * t) & 255;
        float th = (float)ph * STEP;
        IGr[t * 24 + c] = cosf(th) * (1.0f / 256.0f);
        IGi[t * 24 + c] = sinf(th) * (1.0f / 256.0f);
    }
    // Stage D B-matrix (real irfft): x = Tr*ICr + Ti*ICi,
    // ICr = c_ky*cos/256, ICi = -c_ky*sin/256 ; c_0=1, c_{1..11}=2, pad 0.
    for (int ky = 0; ky < 16; ++ky) {
        float coef = (ky == 0) ? 1.0f : ((ky < 12) ? 2.0f : 0.0f);
        int ph = (ky * t) & 255;
        float th = (float)ph * STEP;
        int idx = ((ky >> 1) * 256 + t) * 2 + (ky & 1);
        ICr[idx] = coef * cosf(th) * (1.0f / 256.0f);
        ICi[idx] = -coef * sinf(th) * (1.0f / 256.0f);
    }
}

// ---------------------------------------------------------------------------
// Lifting: h0[b,c,y,x] = pw[c,0]*x + pw[c,1]*gy + pw[c,2]*gx + pb[c]
// ---------------------------------------------------------------------------
__global__ __launch_bounds__(256) void lift_kernel(
        const float* __restrict__ x, const float* __restrict__ pw,
        const float* __restrict__ pb, float* __restrict__ h0) {
    int tid = blockIdx.x * 256 + threadIdx.x;   // < NB*PIX
    int b = tid >> 16;
    int p = tid & 65535;
    int y = p >> 8, xx = p & 255;
    float xv = x[tid];
    float gy = -1.0f + (2.0f / 255.0f) * (float)y;
    float gx = -1.0f + (2.0f / 255.0f) * (float)xx;
#pragma unroll
    for (int c = 0; c < DV; ++c) {
        float v = pw[c * 3 + 0] * xv + pw[c * 3 + 1] * gy + pw[c * 3 + 2] * gx + pb[c];
        h0[(size_t)(b * DV + c) * PIX + p] = v;
    }
}

// ---------------------------------------------------------------------------
// Stage A: T1[row=(b,c,h)][ky] = sum_w h[row][w] * F[w][ky]   (real x complex)
// GEMM M=131072 N=16 K=256. Basis staged in LDS. Output in Stage-B B-frag
// (pair-interleaved over h) layout.
// ---------------------------------------------------------------------------
__global__ __launch_bounds__(128) void stageA_kernel(
        const float* __restrict__ hin,
        const float* __restrict__ Fpr, const float* __restrict__ Fpi,
        float* __restrict__ T1pr, float* __restrict__ T1pi) {
    __shared__ float lds[8192];   // 32 KB: [0..4095]=real pairs, [4096..]=imag pairs
    for (int i = threadIdx.x; i < 4096; i += 128) {
        lds[i] = Fpr[i];
        lds[4096 + i] = Fpi[i];
    }
    __syncthreads();

    int lane = threadIdx.x & 31, wid = threadIdx.x >> 5;
    int gw = blockIdx.x * 4 + wid;
    int rowBase = gw * 16;                 // 16 rows, same (b,c)
    int half = lane >> 4, m = lane & 15;
    const float* arow = hin + (size_t)(rowBase + m) * 256 + 2 * half;

    v8f cr = {}; v8f ci = {};
#pragma unroll 8
    for (int t = 0; t < 64; ++t) {
        v2f a = *(const v2f*)(arow + 4 * t);
        int p = 2 * t + half;
        int boff = (p * 16 + m) * 2;
        v2f br = *(const v2f*)(&lds[boff]);
        v2f bi = *(const v2f*)(&lds[4096 + boff]);
        cr = wmma_f32(a, br, cr);
        ci = wmma_f32(a, bi, ci);
    }
    int bc = rowBase >> 8;
    int hbase = rowBase & 255;
    size_t ob = (size_t)bc * 4096;
#pragma unroll
    for (int v = 0; v < 8; ++v) {
        int h = hbase + v + 8 * half;
        int idx = ((h >> 1) * 16 + m) * 2 + (h & 1);   // pair-interleave over h
        T1pr[ob + idx] = cr[v];
        T1pi[ob + idx] = ci[v];
    }
}

// ---------------------------------------------------------------------------
// Stage B: T2[bc][kx][ky] = sum_h Gh[kx][h] * T1[bc][h][ky]  (complex x complex)
// Per (bc, mtile): M=16 N=16 K=256.
// ---------------------------------------------------------------------------
__global__ __launch_bounds__(128) void stageB_kernel(
        const float* __restrict__ T1pr, const float* __restrict__ T1pi,
        const float* __restrict__ Ghr, const float* __restrict__ Ghi,
        float* __restrict__ T2r, float* __restrict__ T2i) {
    int lane = threadIdx.x & 31, wid = threadIdx.x >> 5;
    int gw = blockIdx.x * 4 + wid;
    int bc = gw >> 1, mt = gw & 1;
    int half = lane >> 4, m = lane & 15;

    const float* arp = Ghr + (size_t)(mt * 16 + m) * 256 + 2 * half;
    const float* aip = Ghi + (size_t)(mt * 16 + m) * 256 + 2 * half;
    const float* brp = T1pr + (size_t)bc * 4096 + m * 2;
    const float* bip = T1pi + (size_t)bc * 4096 + m * 2;

    v8f cr = {}; v8f ci = {};
#pragma unroll 8
    for (int t = 0; t < 64; ++t) {
        v2f ar = *(const v2f*)(arp + 4 * t);
        v2f ai = *(const v2f*)(aip + 4 * t);
        v2f ain = -ai;                         // f32 WMMA has no A-negate modifier
        int p = 2 * t + half;
        v2f br = *(const v2f*)(brp + p * 32);
        v2f bi = *(const v2f*)(bip + p * 32);
        cr = wmma_f32(ain, bi, cr);
        cr = wmma_f32(ar, br, cr);
        ci = wmma_f32(ar, bi, ci);
        ci = wmma_f32(ai, br, ci);
    }
    size_t ob = (size_t)bc * 512;
#pragma unroll
    for (int v = 0; v < 8; ++v) {
        int kx = mt * 16 + v + 8 * half;
        T2r[ob + kx * 16 + m] = cr[v];
        T2i[ob + kx * 16 + m] = ci[v];
    }
}

// ---------------------------------------------------------------------------
// Mode mixing (38 MFLOP -> VALU): out[b,o,kx,ky] = sum_i T2[b,i,kx,ky]*W[i,o,..]
// Writes Stage-C B-frag layout (pair-interleaved over kx), zero-fills ky pads.
// ---------------------------------------------------------------------------
__global__ __launch_bounds__(256) void modemix_kernel(
        const float* __restrict__ T2r, const float* __restrict__ T2i,
        const float* __restrict__ w1r, const float* __restrict__ w1i,
        const float* __restrict__ w2r, const float* __restrict__ w2i,
        float* __restrict__ T3pr, float* __restrict__ T3pi) {
    int tid = blockIdx.x * 256 + threadIdx.x;   // < 16*32*24*16
    int ky = tid & 15;
    int r = tid >> 4;
    int kx = r % 24; r /= 24;
    int o = r & 31;
    int b = r >> 5;
    size_t ob = (size_t)(b * 32 + o) * 384;
    int oidx = ((kx >> 1) * 16 + ky) * 2 + (kx & 1);
    if (ky >= 12) { T3pr[ob + oidx] = 0.0f; T3pi[ob + oidx] = 0.0f; return; }

    const float* wr; const float* wi; int kq;
    if (kx < 12) { wr = w1r; wi = w1i; kq = kx; }
    else         { wr = w2r; wi = w2i; kq = kx - 12; }

    float sr = 0.0f, si = 0.0f;
#pragma unroll 4
    for (int i = 0; i < 32; ++i) {
        float ar = T2r[(size_t)(b * 32 + i) * 512 + kx * 16 + ky];
        float ai = T2i[(size_t)(b * 32 + i) * 512 + kx * 16 + ky];
        size_t widx = ((size_t)(i * 32 + o) * 12 + kq) * 12 + ky;
        float pr = wr[widx], pi = wi[widx];
        sr += ar * pr - ai * pi;
        si += ar * pi + ai * pr;
    }
    T3pr[ob + oidx] = sr;
    T3pi[ob + oidx] = si;
}

// ---------------------------------------------------------------------------
// Conv1x1 skip: x2[b,o,p] = sum_i W[o][i]*h[b,i,p] + bias[o]  -> written to hout
// Per (b, mtile, ntile): M=16 N=16 K=32.
// ---------------------------------------------------------------------------
__global__ __launch_bounds__(128) void conv_skip_kernel(
        const float* __restrict__ hin, const float* __restrict__ wsw,
        const float* __restrict__ wsb, float* __restrict__ hout) {
    int lane = threadIdx.x & 31, wid = threadIdx.x >> 5;
    int gw = blockIdx.x * 4 + wid;           // < 16*2*4096
    int nt = gw & 4095;
    int mt = (gw >> 12) & 1;
    int b  = gw >> 13;
    int half = lane >> 4, m = lane & 15;
    int pbase = nt * 16;
    const float* hb = hin + (size_t)b * DV * PIX + pbase + m;

    v8f c = {};
#pragma unroll
    for (int t = 0; t < 8; ++t) {
        int k0 = 4 * t;
        v2f a = *(const v2f*)(wsw + (mt * 16 + m) * 32 + k0 + 2 * half);
        int kk = k0 + 2 * half;
        v2f bv;
        bv.x = hb[(size_t)kk * PIX];
        bv.y = hb[(size_t)(kk + 1) * PIX];
        c = wmma_f32(a, bv, c);
    }
#pragma unroll
    for (int v = 0; v < 8; ++v) {
        int o = mt * 16 + v + 8 * half;
        hout[(size_t)(b * DV + o) * PIX + pbase + m] = c[v] + wsb[o];
    }
}

// ---------------------------------------------------------------------------
// Stage C: T4[bc][h][ky] = sum_kx IGh[h][kx] * T3[bc][kx][ky] (complex)
// Per (bc, h-tile): M=16 N=16 K=24.
// ---------------------------------------------------------------------------
__global__ __launch_bounds__(128) void stageC_kernel(
        const float* __restrict__ T3pr, const float* __restrict__ T3pi,
        const float* __restrict__ IGr, const float* __restrict__ IGi,
        float* __restrict__ T4r, float* __restrict__ T4i) {
    int lane = threadIdx.x & 31, wid = threadIdx.x >> 5;
    int gw = blockIdx.x * 4 + wid;           // < 512*16
    int bc = gw >> 4;
    int mt = gw & 15;
    int half = lane >> 4, m = lane & 15;
    int hbase = mt * 16;

    const float* arp = IGr + (size_t)(hbase + m) * 24 + 2 * half;
    const float* aip = IGi + (size_t)(hbase + m) * 24 + 2 * half;
    const float* brp = T3pr + (size_t)bc * 384 + m * 2;
    const float* bip = T3pi + (size_t)bc * 384 + m * 2;

    v8f cr = {}; v8f ci = {};
#pragma unroll
    for (int t = 0; t < 6; ++t) {
        v2f ar = *(const v2f*)(arp + 4 * t);
        v2f ai = *(const v2f*)(aip + 4 * t);
        v2f ain = -ai;
        int p = 2 * t + half;
        v2f br = *(const v2f*)(brp + p * 32);
        v2f bi = *(const v2f*)(bip + p * 32);
        cr = wmma_f32(ain, bi, cr);
        cr = wmma_f32(ar, br, cr);
        ci = wmma_f32(ar, bi, ci);
        ci = wmma_f32(ai, br, ci);
    }
    size_t rb = (size_t)bc * 256 + hbase;
#pragma unroll
    for (int v = 0; v < 8; ++v) {
        size_t row = rb + v + 8 * half;
        T4r[row * 16 + m] = cr[v];
        T4i[row * 16 + m] = ci[v];
    }
}

// ---------------------------------------------------------------------------
// Stage D: x1[row][w] = sum_ky T4r*ICr + T4i*ICi ; fused x1 + x2 (+ ReLU).
// Per (row-tile, w-tile): M=16 N=16 K=16.
// ---------------------------------------------------------------------------
__global__ __launch_bounds__(128) void stageD_kernel(
        const float* __restrict__ T4r, const float* __restrict__ T4i,
        const float* __restrict__ ICr, const float* __restrict__ ICi,
        float* __restrict__ hout, int relu) {
    int lane = threadIdx.x & 31, wid = threadIdx.x >> 5;
    int gw = blockIdx.x * 4 + wid;           // < 8192*16
    int mt = gw >> 4;
    int nt = gw & 15;
    int rowBase = mt * 16;
    int half = lane >> 4, m = lane & 15;

    const float* arp = T4r + (size_t)(rowBase + m) * 16 + 2 * half;
    const float* aip = T4i + (size_t)(rowBase + m) * 16 + 2 * half;
    int w0 = nt * 16 + m;
    const float* brp = ICr + w0 * 2;
    const float* bip = ICi + w0 * 2;

    v8f c = {};
#pragma unroll
    for (int t = 0; t < 4; ++t) {
        v2f ar = *(const v2f*)(arp + 4 * t);
        v2f ai = *(const v2f*)(aip + 4 * t);
        int p = 2 * t + half;
        v2f br = *(const v2f*)(brp + p * 512);
        v2f bi = *(const v2f*)(bip + p * 512);
        c = wmma_f32(ar, br, c);
        c = wmma_f32(ai, bi, c);
    }
#pragma unroll
    for (int v = 0; v < 8; ++v) {
        size_t row = (size_t)rowBase + v + 8 * half;
        size_t idx = row * 256 + nt * 16 + m;
        float val = c[v] + hout[idx];        // x1 + conv skip
        if (relu) val = fmaxf(val, 0.0f);
        hout[idx] = val;
    }
}

// ---------------------------------------------------------------------------
// Projection: out[b,y,x] = sum_i q_w[i]*h[b,i,y,x] + q_b
// ---------------------------------------------------------------------------
__global__ __launch_bounds__(256) void project_kernel(
        const float* __restrict__ hin, const float* __restrict__ qw,
        const float* __restrict__ qb, float* __restrict__ out) {
    int tid = blockIdx.x * 256 + threadIdx.x;   // < NB*PIX
    int b = tid >> 16;
    int p = tid & 65535;
    const float* hp = hin + (size_t)b * DV * PIX + p;
    float s = qb[0];
#pragma unroll
    for (int i = 0; i < DV; ++i) s += qw[i] * hp[(size_t)i * PIX];
    out[tid] = s;
}

// ---------------------------------------------------------------------------
extern "C" void kernel_launch(void* const* d_in, const int* in_sizes, int n_in,
                              void* d_out, int out_size, void* d_ws, size_t ws_size,
                              hipStream_t stream) {
    (void)in_sizes; (void)n_in; (void)out_size; (void)ws_size;
    const float* x    = (const float*)d_in[0];
    const float* p_w  = (const float*)d_in[1];
    const float* p_b  = (const float*)d_in[2];
    const float* ws_w = (const float*)d_in[3];   // [L][32][32]
    const float* ws_b = (const float*)d_in[4];   // [L][32]
    const float* w1r  = (const float*)d_in[5];   // [L][32][32][12][12]
    const float* w1i  = (const float*)d_in[6];
    const float* w2r  = (const float*)d_in[7];
    const float* w2i  = (const float*)d_in[8];
    const float* q_w  = (const float*)d_in[9];
    const float* q_b  = (const float*)d_in[10];

    float* ws   = (float*)d_ws;
    float* h0   = ws + OFF_H0;
    float* h1   = ws + OFF_H1;
    float* T1pr = ws + OFF_T1R;  float* T1pi = ws + OFF_T1I;
    float* T2r  = ws + OFF_T2R;  float* T2i  = ws + OFF_T2I;
    float* T3pr = ws + OFF_T3R;  float* T3pi = ws + OFF_T3I;
    float* T4r  = ws + OFF_T4R;  float* T4i  = ws + OFF_T4I;
    float* Fpr  = ws + OFF_FPR;  float* Fpi  = ws + OFF_FPI;
    float* Ghr  = ws + OFF_GHR;  float* Ghi  = ws + OFF_GHI;
    float* IGr  = ws + OFF_IGR;  float* IGi  = ws + OFF_IGI;
    float* ICr  = ws + OFF_ICR;  float* ICi  = ws + OFF_ICI;

    build_basis<<<1, 256, 0, stream>>>(ws);
    lift_kernel<<<(NB * PIX) / 256, 256, 0, stream>>>(x, p_w, p_b, h0);

    constexpr size_t WSTRIDE = 32 * 32 * 12 * 12;   // 147456 per layer
    float* hin  = h0;
    float* hout = h1;
    for (int l = 0; l < NL; ++l) {
        stageA_kernel<<<NROW / 16 / 4, 128, 0, stream>>>(hin, Fpr, Fpi, T1pr, T1pi);
        stageB_kernel<<<BC * 2 / 4, 128, 0, stream>>>(T1pr, T1pi, Ghr, Ghi, T2r, T2i);
        modemix_kernel<<<(NB * 32 * 24 * 16) / 256, 256, 0, stream>>>(
            T2r, T2i, w1r + (size_t)l * WSTRIDE, w1i + (size_t)l * WSTRIDE,
            w2r + (size_t)l * WSTRIDE, w2i + (size_t)l * WSTRIDE, T3pr, T3pi);
        conv_skip_kernel<<<(NB * 2 * (PIX / 16)) / 4, 128, 0, stream>>>(
            hin, ws_w + (size_t)l * 1024, ws_b + (size_t)l * 32, hout);
        stageC_kernel<<<BC * 16 / 4, 128, 0, stream>>>(T3pr, T3pi, IGr, IGi, T4r, T4i);
        stageD_kernel<<<(NROW / 16) * 16 / 4, 128, 0, stream>>>(
            T4r, T4i, ICr, ICi, hout, (l < NL - 1) ? 1 : 0);
        float* tmp = hin; hin = hout; hout = tmp;
    }
    project_kernel<<<(NB * PIX) / 256, 256, 0, stream>>>(hin, q_w, q_b, (float*)d_out);
}